// GCN_137438953659
// MI455X (gfx1250) — compile-verified
//
#include <hip/hip_runtime.h>
#include <hip/hip_bf16.h>

typedef __attribute__((ext_vector_type(2))) float v2f;
typedef __attribute__((ext_vector_type(8))) float v8f;

// Native f32 atomic add (global_atomic_add_f32, no CAS loop).
__device__ __forceinline__ void atomAddF(float* p, float v) {
    unsafeAtomicAdd(p, v);
}

// ---------------------------------------------------------------------------
// init: deg = 1.0 (self loop), agg = 0, pooled output region = 0
// ---------------------------------------------------------------------------
__global__ __launch_bounds__(256) void gcn_init_kernel(float* __restrict__ deg,
                                                       float* __restrict__ agg,
                                                       float* __restrict__ pool,
                                                       int n_nodes, int agg_n, int pool_n) {
    int i = blockIdx.x * 256 + threadIdx.x;
    if (i < agg_n)  agg[i]  = 0.0f;
    if (i < n_nodes) deg[i] = 1.0f;
    if (i < pool_n) pool[i] = 0.0f;
}

// ---------------------------------------------------------------------------
// degree: deg[dst[e]] += 1
// ---------------------------------------------------------------------------
__global__ __launch_bounds__(256) void gcn_degree_kernel(const int* __restrict__ dst,
                                                         float* __restrict__ deg,
                                                         int n_edges) {
    int i = blockIdx.x * 256 + threadIdx.x;
    if (i < n_edges) atomAddF(&deg[dst[i]], 1.0f);
}

// ---------------------------------------------------------------------------
// dinv = rsqrt(deg)   (in place)
// ---------------------------------------------------------------------------
__global__ __launch_bounds__(256) void gcn_rsqrt_kernel(float* __restrict__ deg, int n_nodes) {
    int i = blockIdx.x * 256 + threadIdx.x;
    if (i < n_nodes) deg[i] = __frsqrt_rn(deg[i]);
}

// ---------------------------------------------------------------------------
// H[N,64] = X[N,64] @ W[64,64] via V_WMMA_F32_16X16X4_F32.
// One wave computes a 16x64 output tile: 4 accumulators (16x16 each),
// 16 K-steps of K=4.  W staged in LDS (16KB) per block.
//
// A-fragment (32-bit A 16x4): lane<16 holds rows M=lane, {K=k0,k0+1};
//                             lane>=16 holds rows M=lane-16, {K=k0+2,k0+3}.
// B-fragment (4x16):          VGPR0 = rows K=k0 / k0+2 split at lane 16,
//                             VGPR1 = rows K=k0+1 / k0+3, N = lane&15.
// C/D: VGPR j = row M=j (lanes 0-15), M=j+8 (lanes 16-31).
// ---------------------------------------------------------------------------
__global__ __launch_bounds__(256) void gcn_gemm64_kernel(const float* __restrict__ X,
                                                         const float* __restrict__ W,
                                                         float* __restrict__ H,
                                                         int n_rows) {
    __shared__ float Wlds[64 * 64];
    for (int i = threadIdx.x; i < 64 * 64; i += 256) Wlds[i] = W[i];
    __syncthreads();

    const int wave = threadIdx.x >> 5;
    const int lane = threadIdx.x & 31;
    const int tile = blockIdx.x * 8 + wave;         // 16-row tile index
    if (tile * 16 >= n_rows) return;                // wave-uniform guard

    const int rowBase = tile * 16;
    const int r    = rowBase + (lane & 15);
    const int koff = (lane < 16) ? 0 : 2;
    const int col  = lane & 15;

    v8f acc0 = {}, acc1 = {}, acc2 = {}, acc3 = {};

    const float* xrow = X + (size_t)r * 64;
    #pragma unroll
    for (int k0 = 0; k0 < 64; k0 += 4) {
        v2f a = *(const v2f*)&xrow[k0 + koff];      // 8B aligned (k0+koff even)

        const float* wr0 = &Wlds[(k0 + koff) * 64];
        const float* wr1 = &Wlds[(k0 + koff + 1) * 64];
        v2f b0, b1, b2, b3;
        b0.x = wr0[col];      b0.y = wr1[col];
        b1.x = wr0[col + 16]; b1.y = wr1[col + 16];
        b2.x = wr0[col + 32]; b2.y = wr1[col + 32];
        b3.x = wr0[col + 48]; b3.y = wr1[col + 48];

        acc0 = __builtin_amdgcn_wmma_f32_16x16x4_f32(false, a, false, b0, (short)0, acc0, false, false);
        acc1 = __builtin_amdgcn_wmma_f32_16x16x4_f32(false, a, false, b1, (short)0, acc1, false, false);
        acc2 = __builtin_amdgcn_wmma_f32_16x16x4_f32(false, a, false, b2, (short)0, acc2, false, false);
        acc3 = __builtin_amdgcn_wmma_f32_16x16x4_f32(false, a, false, b3, (short)0, acc3, false, false);
    }

    const int rowHi = (lane >= 16) ? 8 : 0;
    #pragma unroll
    for (int j = 0; j < 8; ++j) {
        float* out = H + (size_t)(rowBase + j + rowHi) * 64;
        out[col]      = acc0[j];
        out[col + 16] = acc1[j];
        out[col + 32] = acc2[j];
        out[col + 48] = acc3[j];
    }
}

// ---------------------------------------------------------------------------
// edge scatter: agg[dst] += Hin[src] * (dinv[src]*dinv[dst]);  wave per edge.
// ---------------------------------------------------------------------------
__global__ __launch_bounds__(256) void gcn_edge_agg_kernel(const int* __restrict__ src,
                                                           const int* __restrict__ dst,
                                                           const float* __restrict__ dinv,
                                                           const float* __restrict__ Hin,
                                                           float* __restrict__ agg,
                                                           int n_edges) {
    const int e = blockIdx.x * 8 + (threadIdx.x >> 5);
    if (e >= n_edges) return;
    const int lane = threadIdx.x & 31;

    const int s = src[e];
    const int d = dst[e];
    const float norm = dinv[s] * dinv[d];

    v2f hv = *(const v2f*)&Hin[(size_t)s * 64 + lane * 2];
    float* ap = &agg[(size_t)d * 64 + lane * 2];
    atomAddF(ap,     hv.x * norm);
    atomAddF(ap + 1, hv.y * norm);
}

// ---------------------------------------------------------------------------
// finalize: Hout = relu(agg + Hlin*dinv^2 + b); optionally zero agg for the
// next layer; accumulate graph pool into pool[g*128 + colOff + c].
// Wave per node, float2 per lane.
// ---------------------------------------------------------------------------
__global__ __launch_bounds__(256) void gcn_finalize_kernel(const float* __restrict__ Hlin,
                                                           float* __restrict__ agg,
                                                           const float* __restrict__ dinv,
                                                           const float* __restrict__ bias,
                                                           const int* __restrict__ batch,
                                                           float* __restrict__ Hout,
                                                           float* __restrict__ pool,
                                                           int poolColOff,
                                                           int n_nodes,
                                                           int zero_agg) {
    const int n = blockIdx.x * 8 + (threadIdx.x >> 5);
    if (n >= n_nodes) return;
    const int lane = threadIdx.x & 31;
    const int c = lane * 2;

    const float dv = dinv[n];
    const float sw = dv * dv;

    v2f a = *(const v2f*)&agg[(size_t)n * 64 + c];
    v2f h = *(const v2f*)&Hlin[(size_t)n * 64 + c];

    float v0 = fmaxf(a.x + h.x * sw + bias[c],     0.0f);
    float v1 = fmaxf(a.y + h.y * sw + bias[c + 1], 0.0f);

    v2f o; o.x = v0; o.y = v1;
    *(v2f*)&Hout[(size_t)n * 64 + c] = o;

    if (zero_agg) {
        v2f z = {};
        *(v2f*)&agg[(size_t)n * 64 + c] = z;
    }

    float* pp = &pool[(size_t)batch[n] * 128 + poolColOff + c];
    atomAddF(pp,     v0);
    atomAddF(pp + 1, v1);
}

// ---------------------------------------------------------------------------
extern "C" void kernel_launch(void* const* d_in, const int* in_sizes, int n_in,
                              void* d_out, int out_size, void* d_ws, size_t ws_size,
                              hipStream_t stream) {
    const float* x    = (const float*)d_in[0];
    const int*   ei   = (const int*)  d_in[1];   // [2,E] flat: src then dst
    const int*   batch= (const int*)  d_in[2];
    const float* W1   = (const float*)d_in[3];
    const float* b1   = (const float*)d_in[4];
    const float* W2   = (const float*)d_in[5];
    const float* b2   = (const float*)d_in[6];

    const int N = in_sizes[2];          // nodes
    const int E = in_sizes[1] / 2;      // edges
    const int FEATN = N * 64;
    const int poolN = out_size - FEATN; // G * 128

    const int* src = ei;
    const int* dst = ei + E;

    float* out  = (float*)d_out;        // [0, N*64) = h2 ; [N*64, ...) = pooled
    float* pool = out + FEATN;

    // workspace layout (floats)
    float* ws   = (float*)d_ws;
    float* dinv = ws;                                   // N
    size_t np   = ((size_t)N + 511) & ~(size_t)511;
    float* hbuf = ws + np;                              // N*64  (xW / h1W2)
    float* agg  = hbuf + (size_t)N * 64;                // N*64
    float* h1   = agg  + (size_t)N * 64;                // N*64

    const int waveBlocksN = (N + 7) / 8;                // wave per node
    const int waveBlocksE = (E + 7) / 8;                // wave per edge
    const int gemmBlocks  = ((N + 15) / 16 + 7) / 8;    // wave per 16-row tile

    // 1) init deg/agg/pool
    gcn_init_kernel<<<(FEATN + 255) / 256, 256, 0, stream>>>(dinv, agg, pool, N, FEATN, poolN);
    // 2) degree + dinv
    gcn_degree_kernel<<<(E + 255) / 256, 256, 0, stream>>>(dst, dinv, E);
    gcn_rsqrt_kernel<<<(N + 255) / 256, 256, 0, stream>>>(dinv, N);
    // 3) layer 1
    gcn_gemm64_kernel<<<gemmBlocks, 256, 0, stream>>>(x, W1, hbuf, N);
    gcn_edge_agg_kernel<<<waveBlocksE, 256, 0, stream>>>(src, dst, dinv, hbuf, agg, E);
    gcn_finalize_kernel<<<waveBlocksN, 256, 0, stream>>>(hbuf, agg, dinv, b1, batch, h1, pool,
                                                         /*colOff=*/0, N, /*zero_agg=*/1);
    // 4) layer 2
    gcn_gemm64_kernel<<<gemmBlocks, 256, 0, stream>>>(h1, W2, hbuf, N);
    gcn_edge_agg_kernel<<<waveBlocksE, 256, 0, stream>>>(src, dst, dinv, hbuf, agg, E);
    gcn_finalize_kernel<<<waveBlocksN, 256, 0, stream>>>(hbuf, agg, dinv, b2, batch, out, pool,
                                                         /*colOff=*/64, N, /*zero_agg=*/0);
}